// ConsistentHashing_61366492725441
// MI455X (gfx1250) — compile-verified
//
#include <hip/hip_runtime.h>
#include <hip/hip_bf16.h>

typedef __attribute__((ext_vector_type(2))) float v2f;
typedef __attribute__((ext_vector_type(8))) float v8f;

#define N_NODES   500000
#define INPUT_DIM 256
#define PROJ_DIM  64
#define NPAD      (1 << 19) /* 524288 */

// ---------------------------------------------------------------------------
// Kernel 1: X_proj = x @ W.T via V_WMMA_F32_16X16X4_F32, fused row-mean.
// One wave (32 lanes) computes 16 rows of x against all 64 rows of W.
// A (16x4 f32): lanes 0-15 hold {K=k,k+1} for M=lane, lanes 16-31 hold
// {K=k+2,k+3} for M=lane-16  -> one float2 load per lane per K-step.
// B (4x16 f32): mirrored layout over W rows (the N dimension).
// ---------------------------------------------------------------------------
__global__ __launch_bounds__(256) void proj_mean_kernel(
    const float* __restrict__ x, const float* __restrict__ W,
    float* __restrict__ keys, int* __restrict__ vals)
{
    const int lane = threadIdx.x & 31;
    const int waveInBlock = threadIdx.x >> 5;
    const int waveId = blockIdx.x * 8 + waveInBlock;
    const int m0 = waveId * 16;
    if (m0 >= N_NODES) return;          // wave-uniform: EXEC all-1s inside

    const int hf  = lane >> 4;          // 0: K pair {0,1}, 1: K pair {2,3}
    const int l15 = lane & 15;
    const float* xrow = x + (size_t)(m0 + l15) * INPUT_DIM + hf * 2;
    const float* wrow = W + (size_t)l15 * INPUT_DIM + hf * 2;

    v8f acc0 = {}; v8f acc1 = {}; v8f acc2 = {}; v8f acc3 = {};

    #pragma unroll 4
    for (int kk = 0; kk < INPUT_DIM; kk += 4) {
        v2f a  = *(const v2f*)(xrow + kk);
        v2f b0 = *(const v2f*)(wrow + kk + (size_t)0  * 16 * INPUT_DIM);
        v2f b1 = *(const v2f*)(wrow + kk + (size_t)1  * 16 * INPUT_DIM);
        v2f b2 = *(const v2f*)(wrow + kk + (size_t)2  * 16 * INPUT_DIM);
        v2f b3 = *(const v2f*)(wrow + kk + (size_t)3  * 16 * INPUT_DIM);
        acc0 = __builtin_amdgcn_wmma_f32_16x16x4_f32(false, a, false, b0, (short)0, acc0, false, false);
        acc1 = __builtin_amdgcn_wmma_f32_16x16x4_f32(false, a, false, b1, (short)0, acc1, false, false);
        acc2 = __builtin_amdgcn_wmma_f32_16x16x4_f32(false, a, false, b2, (short)0, acc2, false, false);
        acc3 = __builtin_amdgcn_wmma_f32_16x16x4_f32(false, a, false, b3, (short)0, acc3, false, false);
    }

    // C/D layout: VGPR r, lane L -> element (M = r + (L<16?0:8), N = L&15).
    // Sum the four N-tiles element-wise, then reduce across the 16 lanes of
    // each half to get the full row sum over N=0..63.
    float s[8];
    #pragma unroll
    for (int r = 0; r < 8; ++r)
        s[r] = acc0[r] + acc1[r] + acc2[r] + acc3[r];

    #pragma unroll
    for (int r = 0; r < 8; ++r) {
        float t = s[r];
        t += __shfl_xor(t, 1, 16);
        t += __shfl_xor(t, 2, 16);
        t += __shfl_xor(t, 4, 16);
        t += __shfl_xor(t, 8, 16);
        s[r] = t;
    }

    if (l15 == 0) {                     // lanes 0 (M=m0+r) and 16 (M=m0+8+r)
        const int M = m0 + hf * 8;
        #pragma unroll
        for (int r = 0; r < 8; ++r) {
            keys[M + r] = s[r] * (1.0f / (float)PROJ_DIM);
            vals[M + r] = M + r;
        }
    }
}

// Pad [N_NODES, NPAD) with +INF keys so real data sorts to the front.
__global__ void pad_kernel(float* __restrict__ keys, int* __restrict__ vals)
{
    int i = blockIdx.x * blockDim.x + threadIdx.x + N_NODES;
    if (i < NPAD) { keys[i] = __builtin_inff(); vals[i] = 0x7fffffff; }
}

// Global bitonic compare-exchange step (j > 256).
__global__ __launch_bounds__(256) void bitonic_step(
    float* __restrict__ keys, int* __restrict__ vals, int j, int k)
{
    int i = blockIdx.x * blockDim.x + threadIdx.x;
    int p = i ^ j;
    if (p > i) {
        float ki = keys[i], kp = keys[p];
        bool up = ((i & k) == 0);
        if ((ki > kp) == up) {
            keys[i] = kp; keys[p] = ki;
            int vi = vals[i]; vals[i] = vals[p]; vals[p] = vi;
        }
    }
}

// LDS bitonic: runs all steps j = jstart..1 (jstart <= 256) for one k-stage.
// Each 256-thread block owns 512 consecutive elements.
__global__ __launch_bounds__(256) void bitonic_lds(
    float* __restrict__ keys, int* __restrict__ vals, int jstart, int k)
{
    __shared__ float sk[512];
    __shared__ int   sv[512];
    const int base = blockIdx.x * 512;
    const int t = threadIdx.x;

    sk[t]       = keys[base + t];       sv[t]       = vals[base + t];
    sk[t + 256] = keys[base + t + 256]; sv[t + 256] = vals[base + t + 256];
    __syncthreads();

    for (int j = jstart; j >= 1; j >>= 1) {
        int i = ((t & ~(j - 1)) << 1) | (t & (j - 1));
        int p = i | j;
        bool up = (((base + i) & k) == 0);
        float ki = sk[i], kp = sk[p];
        if ((ki > kp) == up) {
            sk[i] = kp; sk[p] = ki;
            int vi = sv[i]; sv[i] = sv[p]; sv[p] = vi;
        }
        __syncthreads();
    }

    keys[base + t]       = sk[t];       vals[base + t]       = sv[t];
    keys[base + t + 256] = sk[t + 256]; vals[base + t + 256] = sv[t + 256];
}

// Sorted values -> d_out[0..N); rank scattered to d_out[N..2N) as float.
__global__ void finalize_kernel(
    const float* __restrict__ keys, const int* __restrict__ vals,
    float* __restrict__ out_pos, float* __restrict__ out_idx)
{
    int r = blockIdx.x * blockDim.x + threadIdx.x;
    if (r < N_NODES) {
        out_pos[r] = keys[r];
        out_idx[vals[r]] = (float)r;
    }
}

extern "C" void kernel_launch(void* const* d_in, const int* in_sizes, int n_in,
                              void* d_out, int out_size, void* d_ws, size_t ws_size,
                              hipStream_t stream)
{
    const float* x = (const float*)d_in[0];   // [500000, 256] f32
    const float* W = (const float*)d_in[1];   // [64, 256] f32

    float* keys = (float*)d_ws;                                   // NPAD f32
    int*   vals = (int*)((char*)d_ws + (size_t)NPAD * sizeof(float)); // NPAD i32
    float* out  = (float*)d_out;

    // 1) Projection + mean (WMMA). 31250 waves of 16 rows, 8 waves/block.
    const int nwaves  = (N_NODES + 15) / 16;
    const int nblocks = (nwaves + 7) / 8;
    proj_mean_kernel<<<nblocks, 256, 0, stream>>>(x, W, keys, vals);

    // 2) Pad to power of two.
    const int padN = NPAD - N_NODES;
    pad_kernel<<<(padN + 255) / 256, 256, 0, stream>>>(keys, vals);

    // 3) Bitonic sort of (key,val) pairs: global steps for j>256, then one
    //    LDS kernel covering all j<=256 steps of each stage.
    for (int k = 2; k <= NPAD; k <<= 1) {
        int j = k >> 1;
        for (; j > 256; j >>= 1)
            bitonic_step<<<NPAD / 256, 256, 0, stream>>>(keys, vals, j, k);
        bitonic_lds<<<NPAD / 512, 256, 0, stream>>>(keys, vals, j, k);
    }

    // 4) Emit sorted positions + ranks.
    finalize_kernel<<<(N_NODES + 255) / 256, 256, 0, stream>>>(
        keys, vals, out, out + N_NODES);
}